// YOLO_LOSS_38079180046686
// MI455X (gfx1250) — compile-verified
//
#include <hip/hip_runtime.h>
#include <hip/hip_bf16.h>
#include <stdint.h>

// ---------------- problem constants (from reference) ----------------
#define BATCH        16384
#define GRID_NUM     7
#define CELLS_TOTAL  (BATCH * GRID_NUM * GRID_NUM)   // 802816
#define TPB          256
#define NBLOCKS      ((CELLS_TOTAL + TPB - 1) / TPB) // 3136
#define LAM_COORD    5.0f
#define LAM_NOOBJ    0.5f
#define EPSF         1e-12f
#define IMG_F        448.0f
#define CELL_F       64.0f   // 448 / 7

// ---------------- CDNA5 async global -> LDS copy ----------------
// Per-lane: LDS byte address in a VGPR (vdst), 64-bit global addr in a VGPR
// pair. Tracked by ASYNCcnt. 16B-aligned on both sides at every call site.
__device__ __forceinline__ void async_copy16(uint32_t lds_byte_addr, const void* gaddr) {
  asm volatile("global_load_async_to_lds_b128 %0, %1, off"
               :: "v"(lds_byte_addr), "v"(gaddr)
               : "memory");
}

__device__ __forceinline__ void wait_async_zero() {
#if __has_builtin(__builtin_amdgcn_s_wait_asynccnt)
  __builtin_amdgcn_s_wait_asynccnt(0);
#else
  asm volatile("s_wait_asynccnt 0" ::: "memory");
#endif
}

// Low 32 bits of a generic pointer to __shared__ data == wave-relative LDS
// byte offset (generic = {shared_aperture_hi, lds_offset}).
__device__ __forceinline__ uint32_t lds_offset_of(const void* p) {
  return (uint32_t)(uintptr_t)p;
}

// ---------------- stage 1: per-block partial losses ----------------
__global__ void __launch_bounds__(TPB)
yolo_partial_kernel(const float* __restrict__ y_pre,
                    const float* __restrict__ y_true,
                    float* __restrict__ partials)
{
  __shared__ __align__(16) float s_pre [TPB * 30]; // 30720 B
  __shared__ __align__(16) float s_true[TPB *  8]; //  8192 B
  __shared__ float s_red[TPB / 32];

  const int  tid   = threadIdx.x;
  const long cell0 = (long)blockIdx.x * TPB;
  const float* gp = y_pre  + cell0 * 30;
  const float* gt = y_true + cell0 * 8;

  const uint32_t pre_base  = lds_offset_of(s_pre);
  const uint32_t true_base = lds_offset_of(s_true);

  // y_pre tile: 1920 float4 chunks, coalesced async copies
#pragma unroll
  for (int i = 0; i < 8; ++i) {
    int idx = i * TPB + tid;
    if (idx < (TPB * 30) / 4) {
      async_copy16(pre_base + (uint32_t)idx * 16u, (const void*)(gp + idx * 4));
    }
  }
  // y_true tile: 512 float4 chunks, exactly 2 full rounds
#pragma unroll
  for (int i = 0; i < 2; ++i) {
    int idx = i * TPB + tid;
    async_copy16(true_base + (uint32_t)idx * 16u, (const void*)(gt + idx * 4));
  }

  wait_async_zero();   // this wave's async copies landed in LDS
  __syncthreads();     // all waves' copies visible block-wide

  const float* p = s_pre  + tid * 30;
  const float* t = s_true + tid * 8;

  const float cls = t[0];
  const float tx = t[1], ty = t[2], tw = t[3], th = t[4];

  const float b1x = p[0], b1y = p[1], b1w = p[2], b1h = p[3], c1 = p[4];
  const float b2x = p[5], b2y = p[6], b2w = p[7], b2h = p[8], c2 = p[9];

  // ---- IoU vs target box (target terms hoisted) ----
  const float cxt = tx * CELL_F, cyt = ty * CELL_F;
  const float wt  = tw * IMG_F,  ht  = th * IMG_F;
  const float txmax = cxt + 0.5f * wt, txmin = cxt - 0.5f * wt;
  const float tymax = cyt + 0.5f * ht, tymin = cyt - 0.5f * ht;
  const float tarea = wt * ht;

  auto iou_f = [&](float bx, float by, float bw, float bh) -> float {
    float cxp = bx * CELL_F, cyp = by * CELL_F;
    float wp  = bw * IMG_F,  hp  = bh * IMG_F;
    float iw = fminf(cxp + 0.5f * wp, txmax) - fmaxf(cxp - 0.5f * wp, txmin);
    iw = fmaxf(iw, 0.0f);
    float ih = fminf(cyp + 0.5f * hp, tymax) - fmaxf(cyp - 0.5f * hp, tymin);
    ih = fmaxf(ih, 0.0f);
    float inter = iw * ih;
    return inter / (wp * hp + tarea - inter + EPSF);
  };
  const float iou1 = iou_f(b1x, b1y, b1w, b1h);
  const float iou2 = iou_f(b2x, b2y, b2w, b2h);

  // ---- coord loss ----
  const float swt = sqrtf(tw + EPSF), sht = sqrtf(th + EPSF);
  auto coord_f = [&](float bx, float by, float bw, float bh) -> float {
    float dx = bx - tx, dy = by - ty;
    float dw = sqrtf(bw + EPSF) - swt;
    float dh = sqrtf(bh + EPSF) - sht;
    return dx * dx + dy * dy + dw * dw + dh * dh;
  };

  // ---- class loss: sum((p - onehot)^2) = sum(p^2) + [c valid](1 - 2*p[c]) ----
  const int c = (int)cls - 1;
  float cls_loss = 0.0f;
#pragma unroll
  for (int i = 0; i < 20; ++i) {
    float v = p[10 + i];
    cls_loss = fmaf(v, v, cls_loss);
  }
  if (c >= 0 && c < 20) cls_loss += 1.0f - 2.0f * p[10 + c];

  const float d1 = c1 - iou1, d2 = c2 - iou2;
  const float obj_loss =
      d1 * d1 + d2 * d2 +
      LAM_COORD * (coord_f(b1x, b1y, b1w, b1h) + coord_f(b2x, b2y, b2w, b2h)) +
      cls_loss;
  const float noobj_loss = LAM_NOOBJ * (c1 * c1 + c2 * c2);

  float v = (cls != 0.0f) ? obj_loss : noobj_loss;

  // ---- wave32 tree reduce, then cross-wave via LDS ----
#pragma unroll
  for (int off = 16; off > 0; off >>= 1) v += __shfl_down(v, off, 32);
  if ((tid & 31) == 0) s_red[tid >> 5] = v;
  __syncthreads();
  if (tid == 0) {
    float s = 0.0f;
#pragma unroll
    for (int w = 0; w < TPB / 32; ++w) s += s_red[w];
    partials[blockIdx.x] = s;
  }
}

// ---------------- stage 2: deterministic final reduce ----------------
__global__ void __launch_bounds__(1024)
yolo_reduce_kernel(const float* __restrict__ partials, int n,
                   float* __restrict__ out)
{
  __shared__ float s_red[32];
  float v = 0.0f;
  for (int i = threadIdx.x; i < n; i += blockDim.x) v += partials[i];
#pragma unroll
  for (int off = 16; off > 0; off >>= 1) v += __shfl_down(v, off, 32);
  const int lane = threadIdx.x & 31, wid = threadIdx.x >> 5;
  if (lane == 0) s_red[wid] = v;
  __syncthreads();
  if (threadIdx.x < 32) {
    const int nw = (int)(blockDim.x >> 5);
    float r = (threadIdx.x < (unsigned)nw) ? s_red[threadIdx.x] : 0.0f;
#pragma unroll
    for (int off = 16; off > 0; off >>= 1) r += __shfl_down(r, off, 32);
    if (threadIdx.x == 0) out[0] = r * (1.0f / (float)BATCH);
  }
}

// ---------------- host entry ----------------
extern "C" void kernel_launch(void* const* d_in, const int* in_sizes, int n_in,
                              void* d_out, int out_size, void* d_ws, size_t ws_size,
                              hipStream_t stream) {
  const float* y_pre  = (const float*)d_in[0];
  const float* y_true = (const float*)d_in[1];
  float* partials = (float*)d_ws;   // NBLOCKS floats = 12.25 KB scratch
  float* out      = (float*)d_out;

  yolo_partial_kernel<<<NBLOCKS, TPB, 0, stream>>>(y_pre, y_true, partials);
  yolo_reduce_kernel<<<1, 1024, 0, stream>>>(partials, NBLOCKS, out);
}